// CachingSelfMultiHeadedAttention_39273180955316
// MI455X (gfx1250) — compile-verified
//
#include <hip/hip_runtime.h>
#include <math.h>

#define B_   32
#define DM_  1024
#define H_   16
#define DK_  64
#define TC_  4096
#define TT_  (TC_ + 1)

typedef __attribute__((ext_vector_type(2))) float v2f;
typedef __attribute__((ext_vector_type(8))) float v8f;

// ---------------------------------------------------------------------------
// GEMM: out[32][1024] = X[32][1024] @ W[1024][1024]^T + bias[1024]
// One wave (32 lanes) per 16x16 output tile using V_WMMA_F32_16X16X4_F32.
// grid.x = 128 tiles (2 M-tiles x 64 N-tiles), block = 32 threads (1 wave).
//
// WMMA f32 16x16x4 lane layout (CDNA5 ISA 7.12.2):
//   A (16x4):  lanes 0-15 -> M=lane, K={k0,k0+1};  lanes 16-31 -> M=lane-16, K={k0+2,k0+3}
//   B (4x16):  lanes 0-15 -> N=lane, K={k0,k0+1};  lanes 16-31 -> N=lane-16, K={k0+2,k0+3}
//   C/D (16x16): VGPR r -> M = r (+8 for lanes 16-31), N = lane&15
// Since out = X @ W^T, B[k][n] = W[n][k]: both A and B are contiguous float2
// loads at [row*DM + k0 + (lane>=16 ? 2 : 0)].
// ---------------------------------------------------------------------------
__global__ __launch_bounds__(32) void gemm_xwT_bias_wmma(
    const float* __restrict__ X, const float* __restrict__ W,
    const float* __restrict__ bias, float* __restrict__ out)
{
  const int tile = blockIdx.x;
  const int mb   = (tile >> 6) << 4;   // 0 or 16
  const int nb   = (tile & 63) << 4;   // 0..1008
  const int lane = threadIdx.x;        // 0..31
  const int half = lane >> 4;          // 0 or 1
  const int l15  = lane & 15;

  const float* xrow = X + (size_t)(mb + l15) * DM_;  // A-row for this lane
  const float* wrow = W + (size_t)(nb + l15) * DM_;  // B-row (W row n) for this lane
  const int koff = half * 2;

  v8f c = {};
  #pragma unroll 4
  for (int k0 = 0; k0 < DM_; k0 += 4) {
    v2f a = *(const v2f*)(xrow + k0 + koff);
    v2f b = *(const v2f*)(wrow + k0 + koff);
    c = __builtin_amdgcn_wmma_f32_16x16x4_f32(
        /*neg_a=*/false, a, /*neg_b=*/false, b,
        /*c_mod=*/(short)0, c, /*reuse_a=*/false, /*reuse_b=*/false);
  }

  const float bn = bias[nb + l15];
  #pragma unroll
  for (int r = 0; r < 8; ++r) {
    out[(size_t)(mb + half * 8 + r) * DM_ + (nb + l15)] = c[r] + bn;
  }
}

// ---------------------------------------------------------------------------
// Flash-decode attention for QLEN=1.
// One workgroup per (b, h): 512 workgroups x 256 threads (8 waves).
// Pass 1: scores[t] = (q . k_t) / 8 into LDS (K stream, coalesced 256B rows).
// Workgroup max + sum reductions, then pass 2: out[d] = sum_t p_t * V[t][d]
// with threads laid out so consecutive lanes read consecutive V columns.
// Row TC_ (t == 4096) is the freshly projected k_new / v_new (KV-cache append).
// ---------------------------------------------------------------------------
__global__ __launch_bounds__(256) void attn_decode(
    const float* __restrict__ Q,    // [32][1024] projected query
    const float* __restrict__ Kn,   // [32][1024] new key
    const float* __restrict__ Vn,   // [32][1024] new value
    const float* __restrict__ Kc,   // [32][16][4096][64] key cache
    const float* __restrict__ Vc,   // [32][16][4096][64] value cache
    float* __restrict__ Xout)       // [32][1024] attention output
{
  __shared__ float qs[DK_];
  __shared__ float sc[TT_];
  __shared__ float red[256];
  __shared__ float pv[4 * DK_];

  const int bh  = blockIdx.x;
  const int b   = bh >> 4;
  const int h   = bh & 15;
  const int tid = threadIdx.x;

  if (tid < DK_) qs[tid] = Q[(size_t)b * DM_ + h * DK_ + tid];
  __syncthreads();

  const float* kbase = Kc + ((size_t)(b * H_ + h)) * TC_ * DK_;
  const float* vbase = Vc + ((size_t)(b * H_ + h)) * TC_ * DK_;
  const float* knew  = Kn + (size_t)b * DM_ + h * DK_;
  const float* vnew  = Vn + (size_t)b * DM_ + h * DK_;
  const float scale  = 0.125f;  // 1/sqrt(DK)

  // ---- Pass 1: scores + local max ----
  float lmax = -3.402823466e38f;
  for (int t = tid; t < TT_; t += 256) {
    const float* kr = (t < TC_) ? (kbase + (size_t)t * DK_) : knew;
    if (t + 256 < TC_) __builtin_prefetch(kbase + (size_t)(t + 256) * DK_, 0, 0);
    float dot = 0.f;
    #pragma unroll
    for (int d = 0; d < DK_; d += 4) {
      float4 kv = *(const float4*)(kr + d);
      dot += qs[d] * kv.x + qs[d + 1] * kv.y + qs[d + 2] * kv.z + qs[d + 3] * kv.w;
    }
    float s = dot * scale;
    sc[t] = s;
    lmax = fmaxf(lmax, s);
  }
  red[tid] = lmax;
  __syncthreads();
  #pragma unroll
  for (int s = 128; s > 0; s >>= 1) {
    if (tid < s) red[tid] = fmaxf(red[tid], red[tid + s]);
    __syncthreads();
  }
  const float gmax = red[0];
  __syncthreads();

  // ---- exp + local sum ----
  float lsum = 0.f;
  for (int t = tid; t < TT_; t += 256) {
    float e = __expf(sc[t] - gmax);
    sc[t] = e;
    lsum += e;
  }
  red[tid] = lsum;
  __syncthreads();
  #pragma unroll
  for (int s = 128; s > 0; s >>= 1) {
    if (tid < s) red[tid] += red[tid + s];
    __syncthreads();
  }
  const float inv = 1.f / red[0];

  // ---- Pass 2: attn-weighted V, coalesced across d ----
  const int d = tid & 63;
  const int g = tid >> 6;  // 0..3
  float acc = 0.f;
  for (int t = g; t < TT_; t += 4) {
    const float* vr = (t < TC_) ? (vbase + (size_t)t * DK_) : vnew;
    acc += sc[t] * vr[d];
  }
  pv[tid] = acc;
  __syncthreads();
  if (tid < DK_) {
    float o = (pv[tid] + pv[DK_ + tid] + pv[2 * DK_ + tid] + pv[3 * DK_ + tid]) * inv;
    Xout[(size_t)b * DM_ + h * DK_ + tid] = o;
  }
}

// ---------------------------------------------------------------------------
extern "C" void kernel_launch(void* const* d_in, const int* in_sizes, int n_in,
                              void* d_out, int out_size, void* d_ws, size_t ws_size,
                              hipStream_t stream) {
  (void)in_sizes; (void)n_in; (void)out_size; (void)ws_size;

  const float* q  = (const float*)d_in[0];   // [32,1,1024]
  const float* kc = (const float*)d_in[1];   // [32,16,4096,64]
  const float* vc = (const float*)d_in[2];   // [32,16,4096,64]
  const float* wq = (const float*)d_in[3];
  const float* bq = (const float*)d_in[4];
  const float* wk = (const float*)d_in[5];
  const float* bk = (const float*)d_in[6];
  const float* wv = (const float*)d_in[7];
  const float* bv = (const float*)d_in[8];
  const float* wo = (const float*)d_in[9];
  const float* bo = (const float*)d_in[10];

  float* ws   = (float*)d_ws;
  float* qbuf = ws;                    // 32*1024
  float* kbuf = ws + 1 * B_ * DM_;     // 32*1024
  float* vbuf = ws + 2 * B_ * DM_;     // 32*1024
  float* xbuf = ws + 3 * B_ * DM_;     // 32*1024

  // QKV projections (WMMA f32): out = q @ W^T + b
  gemm_xwT_bias_wmma<<<128, 32, 0, stream>>>(q, wq, bq, qbuf);
  gemm_xwT_bias_wmma<<<128, 32, 0, stream>>>(q, wk, bk, kbuf);
  gemm_xwT_bias_wmma<<<128, 32, 0, stream>>>(q, wv, bv, vbuf);

  // Flash-decode attention over 4096 cached + 1 appended KV rows.
  attn_decode<<<B_ * H_, 256, 0, stream>>>(qbuf, kbuf, vbuf, kc, vc, xbuf);

  // Output projection (WMMA f32): d_out = x @ wo^T + bo
  gemm_xwT_bias_wmma<<<128, 32, 0, stream>>>(xbuf, wo, bo, (float*)d_out);
}